// MultiMarginRankingLoss_13649406067127
// MI455X (gfx1250) — compile-verified
//
#include <hip/hip_runtime.h>
#include <float.h>

typedef __attribute__((ext_vector_type(2))) float v2f;
typedef __attribute__((ext_vector_type(8))) float v8f;

#define NUM_Q 64
#define NUM_P 1024
#define TILE  16
#define NT    (NUM_P / TILE)   // 64 tiles per axis
#define WAVES 8
#define BLOCK (WAVES * 32)     // wave32: 8 waves

// Single-instruction ReLU: v_med3_f32(x, 0, FLT_MAX) == clamp(x, 0, FLT_MAX).
__device__ __forceinline__ float relu1(float x) {
    return __builtin_amdgcn_fmed3f(x, 0.0f, FLT_MAX);
}

// One block per query. Computes
//   qloss[q] = sum_{j:rel} sum_{k:!rel} relu(pred[q,k] - pred[q,j])
// via rank-2 embedding into V_WMMA_F32_16X16X4_F32 tiles:
//   D[m,n] = [rel_m, -rel_m*p_m] . [neg_n*p_n ; neg_n] = rel_m*neg_n*(p_n-p_m)
//
// LDS layout: each 16-pair operand tile is followed by a 16-pair zero tile,
// so lane L of a wave reads index tile*32 + L with NO per-lane select:
// lanes 0-15 get the live K=0,1 columns/rows, lanes 16-31 get the zero
// K=2,3 columns/rows the 16x4/4x16 f32 WMMA operand layout requires.
__global__ __launch_bounds__(BLOCK)
void margin_rank_per_query(const float* __restrict__ pred,
                           const float* __restrict__ y,
                           float* __restrict__ qloss)
{
    __shared__ v2f   sA[2 * NUM_P];   // (rel_j, -rel_j*p_j) | zeros, interleaved per 16
    __shared__ v2f   sB[2 * NUM_P];   // (neg_k*p_k, neg_k)  | zeros, interleaved per 16
    __shared__ float sRed[BLOCK];

    const int q = blockIdx.x;
    const float4* p4 = (const float4*)(pred + q * NUM_P);
    const float4* y4 = (const float4*)(y    + q * NUM_P);

    // Stage: one float4 per thread covers the whole 1024-wide row.
    {
        const int t  = threadIdx.x;            // 0..255
        const int k0 = t * 4;                  // 4 consecutive elems, same 16-tile
        const int i0 = k0 + (k0 & ~15);        // interleaved index of first elem
        float4 pv = p4[t];
        float4 yv = y4[t];
        const float pe[4] = {pv.x, pv.y, pv.z, pv.w};
        const float ye[4] = {yv.x, yv.y, yv.z, yv.w};
        const v2f z = {};
#pragma unroll
        for (int i = 0; i < 4; ++i) {
            float rel = (ye[i] == 1.0f) ? 1.0f : 0.0f;
            float neg = 1.0f - rel;
            v2f a; a.x = rel;         a.y = -rel * pe[i];
            v2f b; b.x = neg * pe[i]; b.y = neg;
            sA[i0 + i]      = a;
            sB[i0 + i]      = b;
            sA[i0 + 16 + i] = z;   // zero half-tile for lanes 16-31
            sB[i0 + 16 + i] = z;
        }
    }
    __syncthreads();

    const int wave = threadIdx.x >> 5;      // wave32
    const int lane = threadIdx.x & 31;

    v8f acc = {};   // per-lane relu(D) accumulator (C/D 8-VGPR layout)

    // Each wave owns J-tiles {wave, wave+8, ...}: NT/WAVES == 8 tiles, uniform.
    for (int jt = wave; jt < NT; jt += WAVES) {
        v2f a = sA[jt * 32 + lane];          // affine, no select
        for (int kt = 0; kt < NT; ++kt) {
            v2f b = sB[kt * 32 + lane];      // affine, no select
            v8f c = {};
            c = __builtin_amdgcn_wmma_f32_16x16x4_f32(
                    /*neg_a=*/false, a, /*neg_b=*/false, b,
                    /*c_mod=*/(short)0, c,
                    /*reuse_a=*/false, /*reuse_b=*/false);
#pragma unroll
            for (int r = 0; r < 8; ++r)
                acc[r] += relu1(c[r]);
        }
    }

    float s = acc[0] + acc[1] + acc[2] + acc[3]
            + acc[4] + acc[5] + acc[6] + acc[7];
    sRed[threadIdx.x] = s;
    __syncthreads();
    for (int off = BLOCK / 2; off > 0; off >>= 1) {
        if (threadIdx.x < off) sRed[threadIdx.x] += sRed[threadIdx.x + off];
        __syncthreads();
    }
    if (threadIdx.x == 0) qloss[q] = sRed[0];
}

// Deterministically (re)writes d_out every call: mean over queries.
__global__ void margin_rank_finalize(const float* __restrict__ qloss,
                                     float* __restrict__ out)
{
    if (threadIdx.x == 0 && blockIdx.x == 0) {
        float s = 0.0f;
        for (int i = 0; i < NUM_Q; ++i) s += qloss[i];
        out[0] = s * (1.0f / (float)NUM_Q);
    }
}

extern "C" void kernel_launch(void* const* d_in, const int* in_sizes, int n_in,
                              void* d_out, int out_size, void* d_ws, size_t ws_size,
                              hipStream_t stream)
{
    (void)in_sizes; (void)n_in; (void)out_size; (void)ws_size;
    const float* pred = (const float*)d_in[0];
    const float* y    = (const float*)d_in[1];
    float* out   = (float*)d_out;
    float* qloss = (float*)d_ws;   // 64 floats of scratch

    margin_rank_per_query<<<NUM_Q, BLOCK, 0, stream>>>(pred, y, qloss);
    margin_rank_finalize<<<1, 32, 0, stream>>>(qloss, out);
}